// PointMixerIntraSetLayerPaper_20899310862381
// MI455X (gfx1250) — compile-verified
//
#include <hip/hip_runtime.h>
#include <stdint.h>

#define EPS_BN 1e-5f

typedef float v2f __attribute__((ext_vector_type(2)));
typedef float v8f __attribute__((ext_vector_type(8)));

__device__ __forceinline__ v8f wmma_f32_k4(v2f a, v2f b, v8f c) {
  // D = A(16x4) * B(4x16) + C(16x16), fp32, wave32
  return __builtin_amdgcn_wmma_f32_16x16x4_f32(false, a, false, b, (short)0, c,
                                               false, false);
}

// ---------------------------------------------------------------- utilities
__global__ void k_zero(float* w, int cnt) {
  int i = blockIdx.x * blockDim.x + threadIdx.x;
  if (i < cnt) w[i] = 0.f;
}

// Per-channel sum / sumsq over rows of src[rows][C].  Launch so that
// gridDim.x*blockDim.x is a multiple of C (we use 768*256 = 196608: %3,%8,%64==0).
__global__ void k_colstats(const float* __restrict__ src, long long rows, int C,
                           float* __restrict__ gsum, float* __restrict__ gsq) {
  __shared__ float ls[64];
  __shared__ float lq[64];
  if (threadIdx.x < (unsigned)C) { ls[threadIdx.x] = 0.f; lq[threadIdx.x] = 0.f; }
  __syncthreads();
  long long total  = rows * (long long)C;
  long long stride = (long long)gridDim.x * blockDim.x;
  long long i0     = (long long)blockIdx.x * blockDim.x + threadIdx.x;
  int c = (int)(i0 % C);   // fixed per thread because stride % C == 0
  float s = 0.f, q = 0.f;
  for (long long i = i0; i < total; i += stride) {
    __builtin_prefetch(src + i + stride, 0, 1);   // global_prefetch_b8
    float v = src[i];
    s += v;
    q += v * v;
  }
  atomicAdd(&ls[c], s);
  atomicAdd(&lq[c], q);
  __syncthreads();
  if (threadIdx.x < (unsigned)C) {
    atomicAdd(&gsum[threadIdx.x], ls[threadIdx.x]);
    atomicAdd(&gsq[threadIdx.x],  lq[threadIdx.x]);
  }
}

// scale = g/sqrt(var+eps), shift = b - mean*scale  (biased var, as reference)
__global__ void k_finalize(const float* __restrict__ gsum, const float* __restrict__ gsq,
                           const float* __restrict__ gamma, const float* __restrict__ beta,
                           float* __restrict__ scale, float* __restrict__ shift,
                           float cnt, int C) {
  int c = threadIdx.x;
  if (c >= C) return;
  float m  = gsum[c] / cnt;
  float v  = gsq[c] / cnt - m * m;
  float sc = gamma[c] * rsqrtf(v + EPS_BN);
  scale[c] = sc;
  shift[c] = beta[c] - m * sc;
}

// ------------------------------------------------- stage 1a: geometry only
__global__ void k_geom(const float* __restrict__ p, const long long* __restrict__ knn,
                       const float* __restrict__ lp_w1, const float* __restrict__ lp_b1,
                       float* __restrict__ out_pr, float* __restrict__ pe_pre,
                       long long nk) {
  long long i = (long long)blockIdx.x * blockDim.x + threadIdx.x;
  if (i >= nk) return;
  long long pt  = i >> 4;
  long long nbr = knn[i];
  float pr0 = p[nbr * 3 + 0] - p[pt * 3 + 0];
  float pr1 = p[nbr * 3 + 1] - p[pt * 3 + 1];
  float pr2 = p[nbr * 3 + 2] - p[pt * 3 + 2];
  out_pr[i * 3 + 0] = pr0;
  out_pr[i * 3 + 1] = pr1;
  out_pr[i * 3 + 2] = pr2;
#pragma unroll
  for (int c = 0; c < 3; ++c)
    pe_pre[i * 3 + c] = lp_b1[c] + pr0 * lp_w1[c * 3 + 0] + pr1 * lp_w1[c * 3 + 1] +
                        pr2 * lp_w1[c * 3 + 2];
}

// ------------------------------------------------- stage 1b: WMMA gather-GEMM
// elin[pt*16+k][o] = relu( [p_r | x[nbr]] (16x67) @ w1^T (67x16) + b1 )
// One wave per point; K padded 67 -> 68 (17 WMMA k-steps).
__global__ void k_elin_wmma(const float* __restrict__ p, const float* __restrict__ x,
                            const long long* __restrict__ knn,
                            const float* __restrict__ w1, const float* __restrict__ b1,
                            float* __restrict__ elin, long long n) {
  __shared__ float sw1[16 * 68];          // zero-padded B stage: sw1[o*68+c]
  for (int idx = threadIdx.x; idx < 16 * 68; idx += blockDim.x) {
    int o = idx / 68, c = idx % 68;
    sw1[idx] = (c < 67) ? w1[o * 67 + c] : 0.f;
  }
  __syncthreads();
  int wave = threadIdx.x >> 5;
  int lane = threadIdx.x & 31;
  int half = lane >> 4;
  int l    = lane & 15;
  long long pt = (long long)blockIdx.x * 8 + wave;
  if (pt >= n) return;                    // wave-uniform
  long long nbr = knn[pt * 16 + l];       // this lane's A-row = neighbor l
  float pr0 = p[nbr * 3 + 0] - p[pt * 3 + 0];
  float pr1 = p[nbr * 3 + 1] - p[pt * 3 + 1];
  float pr2 = p[nbr * 3 + 2] - p[pt * 3 + 2];
  const float* xr = x + nbr * 64;
  v8f acc;
#pragma unroll
  for (int v = 0; v < 8; ++v) acc[v] = b1[l];   // bias per output col o=l
#pragma unroll
  for (int s = 0; s < 68; s += 4) {
    int kk = s + 2 * half;
    float a0, a1;
    if (s == 0) {
      a0 = half ? pr2 : pr0;
      a1 = half ? xr[0] : pr1;            // c=3 -> x[0]
    } else if (s == 64) {
      a0 = xr[kk - 3];                    // c=64 -> x[61]; c=66 -> x[63]
      a1 = half ? 0.f : xr[kk - 2];       // c=67 is pad
    } else {
      a0 = xr[kk - 3];
      a1 = xr[kk - 2];
    }
    v2f a = {a0, a1};
    v2f b = {sw1[l * 68 + kk], sw1[l * 68 + kk + 1]};
    acc = wmma_f32_k4(a, b, acc);
  }
#pragma unroll
  for (int v = 0; v < 8; ++v)
    elin[(pt * 16 + v + 8 * half) * 16 + l] = fmaxf(acc[v], 0.f);
}

// ------------------------------------------------- stage 2: WMMA bilinear
// e2 = G @ B + bb, G[k][i*16+j] = E[k][i]*E[k][j] (built on the fly),
// B[i*16+j][o] = bw[o*256 + i*16+j] (LDS).  One point = one 16x16 tile, K=256.
__global__ void k_bilinear_wmma(const float* __restrict__ elin,
                                const float* __restrict__ bw,
                                const float* __restrict__ bb,
                                float* __restrict__ e2, long long n) {
  __shared__ float sbw[16 * 16 * 16];
  for (int j = threadIdx.x; j < 4096; j += blockDim.x) sbw[j] = bw[j];
  __syncthreads();
  int wave = threadIdx.x >> 5;
  int lane = threadIdx.x & 31;
  int half = lane >> 4;
  int l    = lane & 15;
  long long pt = (long long)blockIdx.x * 8 + wave;
  if (pt >= n) return;                    // wave-uniform
  const float* er = elin + (pt * 16 + l) * 16;   // E row for A-row M=l
  float ev[16];
#pragma unroll
  for (int c = 0; c < 16; ++c) ev[c] = er[c];
  // Hoist the half-dependent j-selects so all indices are compile-time.
  float evp[8];
#pragma unroll
  for (int qi = 0; qi < 4; ++qi) {
    evp[2 * qi + 0] = half ? ev[4 * qi + 2] : ev[4 * qi + 0];
    evp[2 * qi + 1] = half ? ev[4 * qi + 3] : ev[4 * qi + 1];
  }
  v8f acc;
#pragma unroll
  for (int v = 0; v < 8; ++v) acc[v] = bb[l];
#pragma unroll
  for (int s = 0; s < 256; s += 4) {
    const int i  = s >> 4;                // row index of outer product (compile-time)
    const int qi = (s & 15) >> 2;
    int kk = s + 2 * half;
    v2f a = {ev[i] * evp[2 * qi], ev[i] * evp[2 * qi + 1]};
    v2f b = {sbw[l * 256 + kk], sbw[l * 256 + kk + 1]};
    acc = wmma_f32_k4(a, b, acc);
  }
#pragma unroll
  for (int v = 0; v < 8; ++v)
    e2[(pt * 16 + v + 8 * half) * 16 + l] = acc[v];
}

// ------------------------------------------------- stage 3: p_embed -> shrink -> energy
__global__ void k_energy(const float* __restrict__ pe_pre, const float* __restrict__ e2,
                         const float* __restrict__ lp_w2, const float* __restrict__ lp_b2,
                         const float* __restrict__ psc, const float* __restrict__ psh,
                         float* __restrict__ energy, long long nk) {
  long long i = (long long)blockIdx.x * blockDim.x + threadIdx.x;
  if (i >= nk) return;
  float pe[3];
#pragma unroll
  for (int c = 0; c < 3; ++c)
    pe[c] = fmaxf(pe_pre[i * 3 + c] * psc[c] + psh[c], 0.f);
  float shr[16];
#pragma unroll
  for (int j = 0; j < 16; ++j) shr[j] = 0.f;
#pragma unroll
  for (int cc = 0; cc < 64; ++cc) {
    float pv = lp_b2[cc] + pe[0] * lp_w2[cc * 3 + 0] + pe[1] * lp_w2[cc * 3 + 1] +
               pe[2] * lp_w2[cc * 3 + 2];
    shr[cc & 15] += pv;
  }
  float* er = energy + i * 32;
#pragma unroll
  for (int o = 0; o < 16; ++o) er[o] = e2[i * 16 + o];
#pragma unroll
  for (int j = 0; j < 16; ++j) er[16 + j] = shr[j];
}

// ------------------------------------------------- WMMA GEMM: g1[M,64] = energy[M,32] @ c1_w^T
__global__ void k_gemm_g1(const float* __restrict__ A, const float* __restrict__ c1w,
                          float* __restrict__ g1, long long M) {
  int wave = threadIdx.x >> 5;
  int lane = threadIdx.x & 31;
  int half = lane >> 4;
  int l    = lane & 15;
  long long row0 = ((long long)blockIdx.x * 2 + (wave >> 2)) * 16;
  int colb = (wave & 3) * 16;
  if (row0 >= M) return;                 // wave-uniform: EXEC stays full
  const float* arow = A + (row0 + l) * 32;
  const float* brow = c1w + (colb + l) * 32;     // B[k][col] = c1_w[col][k]
  v8f acc = {0.f, 0.f, 0.f, 0.f, 0.f, 0.f, 0.f, 0.f};
#pragma unroll
  for (int k = 0; k < 32; k += 4) {
    int kk = k + 2 * half;
    v2f a = {arow[kk], arow[kk + 1]};
    v2f b = {brow[kk], brow[kk + 1]};
    acc = wmma_f32_k4(a, b, acc);
  }
#pragma unroll
  for (int v = 0; v < 8; ++v)
    g1[(row0 + v + 8 * half) * 64 + colb + l] = acc[v];
}

// ------------------------------------------------- WMMA GEMM: g2[M,8] = relu(bn1(g1)) @ c2_w^T
__global__ void k_gemm_g2(const float* __restrict__ g1, const float* __restrict__ c2w,
                          const float* __restrict__ s1, const float* __restrict__ t1,
                          float* __restrict__ g2, long long M) {
  int wave = threadIdx.x >> 5;
  int lane = threadIdx.x & 31;
  int half = lane >> 4;
  int l    = lane & 15;
  long long row0 = ((long long)blockIdx.x * 8 + wave) * 16;
  if (row0 >= M) return;
  const float* arow = g1 + (row0 + l) * 64;
  const float* brow = c2w + (l & 7) * 64;        // valid ptr for all lanes
  float bm = (l < 8) ? 1.f : 0.f;                // pad cols 8..15 with zeros
  v8f acc = {0.f, 0.f, 0.f, 0.f, 0.f, 0.f, 0.f, 0.f};
#pragma unroll
  for (int k = 0; k < 64; k += 4) {
    int kk = k + 2 * half;
    v2f a = {fmaxf(arow[kk] * s1[kk] + t1[kk], 0.f),
             fmaxf(arow[kk + 1] * s1[kk + 1] + t1[kk + 1], 0.f)};
    v2f b = {brow[kk] * bm, brow[kk + 1] * bm};
    acc = wmma_f32_k4(a, b, acc);
  }
  if (l < 8) {
#pragma unroll
    for (int v = 0; v < 8; ++v)
      g2[(row0 + v + 8 * half) * 8 + l] = acc[v];
  }
}

// ------------------------------------------------- WMMA GEMM: xv[n,64] = x[n,64] @ w3^T + b3
__global__ void k_gemm_xv(const float* __restrict__ x, const float* __restrict__ w3,
                          const float* __restrict__ b3, float* __restrict__ xv,
                          long long n) {
  int wave = threadIdx.x >> 5;
  int lane = threadIdx.x & 31;
  int half = lane >> 4;
  int l    = lane & 15;
  long long row0 = ((long long)blockIdx.x * 2 + (wave >> 2)) * 16;
  int colb = (wave & 3) * 16;
  if (row0 >= n) return;
  long long rl = row0 + l;
  if (rl > n - 1) rl = n - 1;                    // clamp tail loads
  const float* arow = x + rl * 64;
  const float* brow = w3 + (colb + l) * 64;
  float bias = b3[colb + l];
  v8f acc;
#pragma unroll
  for (int v = 0; v < 8; ++v) acc[v] = bias;     // bias depends only on col
#pragma unroll
  for (int k = 0; k < 64; k += 4) {
    int kk = k + 2 * half;
    v2f a = {arow[kk], arow[kk + 1]};
    v2f b = {brow[kk], brow[kk + 1]};
    acc = wmma_f32_k4(a, b, acc);
  }
#pragma unroll
  for (int v = 0; v < 8; ++v) {
    long long r = row0 + v + 8 * half;
    if (r < n) xv[r * 64 + colb + l] = acc[v];
  }
}

// ------------------------------------------------- softmax weights over k (axis=1)
__global__ void k_weights(const float* __restrict__ g2, const float* __restrict__ c3w,
                          const float* __restrict__ c3b, const float* __restrict__ s2,
                          const float* __restrict__ t2, float* __restrict__ wgt,
                          long long n) {
  long long i = (long long)blockIdx.x * blockDim.x + threadIdx.x;
  if (i >= n * 8) return;
  long long pt = i >> 3;
  int j = (int)(i & 7);
  float lg[16];
  float mx = -3.0e38f;
#pragma unroll
  for (int k = 0; k < 16; ++k) {
    const float* gr = g2 + (pt * 16 + k) * 8;
    float s = c3b[j];
#pragma unroll
    for (int c = 0; c < 8; ++c)
      s += fmaxf(gr[c] * s2[c] + t2[c], 0.f) * c3w[j * 8 + c];
    lg[k] = s;
    mx = fmaxf(mx, s);
  }
  float sum = 0.f;
#pragma unroll
  for (int k = 0; k < 16; ++k) { lg[k] = expf(lg[k] - mx); sum += lg[k]; }
  float inv = 1.f / sum;
#pragma unroll
  for (int k = 0; k < 16; ++k) wgt[(pt * 16 + k) * 8 + j] = lg[k] * inv;
}

// ------------------------------------------------- final: xk = (xv[nbr] + p_embed) * w
__global__ void k_final_xk(const float* __restrict__ pe_pre, const float* __restrict__ psc,
                           const float* __restrict__ psh, const float* __restrict__ lp_w2,
                           const float* __restrict__ lp_b2,
                           const long long* __restrict__ knn, const float* __restrict__ xv,
                           const float* __restrict__ wgt, float* __restrict__ xk_out,
                           long long nk) {
  long long i = (long long)blockIdx.x * blockDim.x + threadIdx.x;
  if (i >= nk) return;
  float pe[3];
#pragma unroll
  for (int c = 0; c < 3; ++c)
    pe[c] = fmaxf(pe_pre[i * 3 + c] * psc[c] + psh[c], 0.f);
  long long nbr = knn[i];
  const float* xr = xv + nbr * 64;
  const float* wr = wgt + i * 8;
  float* outr = xk_out + i * 64;
#pragma unroll
  for (int c = 0; c < 64; ++c) {
    float pv = lp_b2[c] + pe[0] * lp_w2[c * 3 + 0] + pe[1] * lp_w2[c * 3 + 1] +
               pe[2] * lp_w2[c * 3 + 2];
    outr[c] = (xr[c] + pv) * wr[c & 7];
  }
}

__global__ void k_xout(const float* __restrict__ xk, float* __restrict__ xout,
                       long long n) {
  long long i = (long long)blockIdx.x * blockDim.x + threadIdx.x;
  if (i >= n * 64) return;
  long long pt = i >> 6;
  int c = (int)(i & 63);
  float s = 0.f;
#pragma unroll
  for (int k = 0; k < 16; ++k) s += xk[(pt * 16 + k) * 64 + c];
  xout[i] = s;
}

__global__ void k_copyknn(const long long* __restrict__ knn, long long* __restrict__ dst,
                          long long nk) {
  long long i = (long long)blockIdx.x * blockDim.x + threadIdx.x;
  if (i < nk) dst[i] = knn[i];
}

// ================================================================ launcher
extern "C" void kernel_launch(void* const* d_in, const int* in_sizes, int n_in,
                              void* d_out, int out_size, void* d_ws, size_t ws_size,
                              hipStream_t stream) {
  const float*     p     = (const float*)d_in[0];
  const float*     x     = (const float*)d_in[1];
  const long long* knn   = (const long long*)d_in[2];
  const float*     w1    = (const float*)d_in[3];
  const float*     b1    = (const float*)d_in[4];
  const float*     bwp   = (const float*)d_in[5];
  const float*     bbp   = (const float*)d_in[6];
  const float*     lp_w1 = (const float*)d_in[7];
  const float*     lp_b1 = (const float*)d_in[8];
  const float*     bnp_g = (const float*)d_in[9];
  const float*     bnp_b = (const float*)d_in[10];
  const float*     lp_w2 = (const float*)d_in[11];
  const float*     lp_b2 = (const float*)d_in[12];
  const float*     c1w   = (const float*)d_in[13];
  const float*     bn1_g = (const float*)d_in[14];
  const float*     bn1_b = (const float*)d_in[15];
  const float*     c2w   = (const float*)d_in[16];
  const float*     bn2_g = (const float*)d_in[17];
  const float*     bn2_b = (const float*)d_in[18];
  const float*     c3w   = (const float*)d_in[19];
  const float*     c3b   = (const float*)d_in[20];
  const float*     w3    = (const float*)d_in[21];
  const float*     b3    = (const float*)d_in[22];

  long long n  = (long long)in_sizes[0] / 3;
  long long nk = n * 16;

  float* ws = (float*)d_ws;
  // stats region (512 floats)
  float* bnp_sum = ws + 0;   float* bnp_sq = ws + 3;
  float* bnp_sc  = ws + 6;   float* bnp_sh = ws + 9;
  float* bn1_sum = ws + 16;  float* bn1_sq = ws + 80;
  float* bn1_sc  = ws + 144; float* bn1_sh = ws + 208;
  float* bn2_sum = ws + 272; float* bn2_sq = ws + 280;
  float* bn2_sc  = ws + 288; float* bn2_sh = ws + 296;
  // staged tensors
  float* pe_pre = ws + 512;
  float* e2     = pe_pre + nk * 3;
  float* energy = e2 + nk * 16;
  float* g1     = energy + nk * 32;
  float* g2     = g1 + nk * 64;
  float* xv     = g2 + nk * 8;
  float* wgt    = xv + n * 64;
  float* elin   = g1;  // temp reuse (g1 region unwritten until k_gemm_g1)

  // d_out layout: [x_out n*64 f32][xk nk*64 f32][knn nk i64][p_r nk*3 f32]
  float*     out_x   = (float*)d_out;
  float*     out_xk  = out_x + n * 64;
  long long* out_knn = (long long*)(out_xk + nk * 64);
  float*     out_pr  = (float*)(out_knn + nk);

  const int T = 256;
  unsigned gk = (unsigned)((nk + T - 1) / T);
  unsigned gp = (unsigned)((n + 7) / 8);   // wave-per-point kernels, 8 waves/block

  k_zero<<<1, 512, 0, stream>>>(ws, 512);

  k_geom<<<gk, T, 0, stream>>>(p, knn, lp_w1, lp_b1, out_pr, pe_pre, nk);
  k_elin_wmma<<<gp, T, 0, stream>>>(p, x, knn, w1, b1, elin, n);
  k_bilinear_wmma<<<gp, T, 0, stream>>>(elin, bwp, bbp, e2, n);

  k_colstats<<<768, 256, 0, stream>>>(pe_pre, nk, 3, bnp_sum, bnp_sq);
  k_finalize<<<1, 64, 0, stream>>>(bnp_sum, bnp_sq, bnp_g, bnp_b, bnp_sc, bnp_sh,
                                   (float)nk, 3);

  k_energy<<<gk, T, 0, stream>>>(pe_pre, e2, lp_w2, lp_b2, bnp_sc, bnp_sh, energy, nk);

  unsigned rt1 = (unsigned)(((nk / 16) + 1) / 2);
  k_gemm_g1<<<rt1, 256, 0, stream>>>(energy, c1w, g1, nk);

  k_colstats<<<768, 256, 0, stream>>>(g1, nk, 64, bn1_sum, bn1_sq);
  k_finalize<<<1, 64, 0, stream>>>(bn1_sum, bn1_sq, bn1_g, bn1_b, bn1_sc, bn1_sh,
                                   (float)nk, 64);

  unsigned rt2 = (unsigned)(((nk / 16) + 7) / 8);
  k_gemm_g2<<<rt2, 256, 0, stream>>>(g1, c2w, bn1_sc, bn1_sh, g2, nk);

  k_colstats<<<768, 256, 0, stream>>>(g2, nk, 8, bn2_sum, bn2_sq);
  k_finalize<<<1, 64, 0, stream>>>(bn2_sum, bn2_sq, bn2_g, bn2_b, bn2_sc, bn2_sh,
                                   (float)nk, 8);

  unsigned rtx = (unsigned)((((n + 15) / 16) + 1) / 2);
  k_gemm_xv<<<rtx, 256, 0, stream>>>(x, w3, b3, xv, n);

  k_weights<<<(unsigned)((n * 8 + T - 1) / T), T, 0, stream>>>(g2, c3w, c3b, bn2_sc,
                                                               bn2_sh, wgt, n);
  k_final_xk<<<gk, T, 0, stream>>>(pe_pre, bnp_sc, bnp_sh, lp_w2, lp_b2, knn, xv, wgt,
                                   out_xk, nk);
  k_xout<<<(unsigned)((n * 64 + T - 1) / T), T, 0, stream>>>(out_xk, out_x, n);
  k_copyknn<<<gk, T, 0, stream>>>(knn, out_knn, nk);
}